// QuantizerDecoder_49727131353591
// MI455X (gfx1250) — compile-verified
//
#include <hip/hip_runtime.h>

typedef __attribute__((ext_vector_type(2))) float v2f;
typedef __attribute__((ext_vector_type(8))) float v8f;

// Problem constants (from reference):
//   codes:    (32, 64, 64, 8) int32
//   codebook: (8, 4096, 64)   f32
//   wv:       (8, 64, 64)     f32
//   v[m,k,c] = sum_d codebook[m,k,d] * wv[m,c,d]      (8,4096,64)
//   out[n, m*64+c, h, w] = v[m, codes[n,h,w,m], c]    (32,512,64,64)

#define MDIM 8
#define KCODES 4096
#define DG 64
#define NB 32
#define HW 64

// ---------------------------------------------------------------------------
// Phase 1: v = codebook @ wv^T per m, via V_WMMA_F32_16X16X4_F32.
// One wave computes one 16x16 tile of v[m]; K-dim (d) = 64 in 16 steps of 4.
// Grid: 8 m * 256 k-tiles * 4 c-tiles = 8192 waves = 1024 blocks x 256 thr.
// ---------------------------------------------------------------------------
__global__ __launch_bounds__(256) void qd_gemm_wmma(
    const float* __restrict__ codebook,
    const float* __restrict__ wv,
    float* __restrict__ v)
{
  const int wave = blockIdx.x * 8 + (threadIdx.x >> 5);
  const int lane = threadIdx.x & 31;

  const int m  = wave >> 10;          // 0..7
  const int kt = (wave >> 2) & 255;   // 0..255
  const int ct = wave & 3;            // 0..3
  const int k0 = kt * 16;
  const int c0 = ct * 16;

  const float* __restrict__ cb  = codebook + ((size_t)m * KCODES + k0) * DG; // 16 rows x 64
  const float* __restrict__ wvm = wv + (size_t)m * DG * DG;                  // [c][d] 64x64

  const int half = lane >> 4;   // 0: lanes 0-15, 1: lanes 16-31
  const int l    = lane & 15;

  v8f acc = {0.f, 0.f, 0.f, 0.f, 0.f, 0.f, 0.f, 0.f};

  #pragma unroll
  for (int d = 0; d < DG; d += 4) {
    const int da = d + 2 * half;
    // A 16x4 (ISA layout): VGPR0 = K even slot, VGPR1 = K odd slot;
    // lanes 0-15 carry K = d..d+1, lanes 16-31 carry K = d+2..d+3.
    v2f a;
    a.x = cb[l * DG + da];
    a.y = cb[l * DG + da + 1];
    // B 4x16: B[k][n] = wv[m][c0+n][k]; mirrored lane striping.
    v2f b;
    b.x = wvm[(c0 + l) * DG + da];
    b.y = wvm[(c0 + l) * DG + da + 1];
    acc = __builtin_amdgcn_wmma_f32_16x16x4_f32(
        /*neg_a=*/false, a, /*neg_b=*/false, b,
        /*c_mod=*/(short)0, acc, /*reuse_a=*/false, /*reuse_b=*/false);
  }

  // D 16x16: VGPR i -> row M = i + 8*half, col N = l.
  float* __restrict__ vout = v + ((size_t)m * KCODES + k0) * DG + c0;
  #pragma unroll
  for (int i = 0; i < 8; ++i) {
    vout[(size_t)(i + 8 * half) * DG + l] = acc[i];
  }
}

// ---------------------------------------------------------------------------
// Phase 2: gather + transpose scatter (HBM-store bound; float4 stores).
// Block = (n, m, h). 64 codes staged in LDS. Thread owns w4 = 4 consecutive
// w's and sweeps c in 4 passes -> 4 scalar L2 gathers + 1 float4 HBM store
// per pass; each half-wave's store covers 256 contiguous bytes.
// Grid: 32*8*64 = 16384 blocks x 256 threads.
// ---------------------------------------------------------------------------
__global__ __launch_bounds__(256) void qd_gather(
    const int* __restrict__ codes,
    const float* __restrict__ v,
    float* __restrict__ out)
{
  const int b = blockIdx.x;
  const int h = b & 63;
  const int m = (b >> 6) & 7;
  const int n = b >> 9;

  __shared__ int s_codes[HW];
  const int t = threadIdx.x;
  if (t < HW) {
    // codes[n][h][w][m]
    s_codes[t] = codes[(((size_t)n * HW + h) * HW + t) * MDIM + m];
  }
  __syncthreads();

  const int w4 = (t & 15) * 4;   // 0,4,...,60
  const int c0 = t >> 4;         // 0..15

  const float* __restrict__ vm = v + (size_t)m * KCODES * DG;
  const float* __restrict__ r0 = vm + (size_t)s_codes[w4 + 0] * DG;
  const float* __restrict__ r1 = vm + (size_t)s_codes[w4 + 1] * DG;
  const float* __restrict__ r2 = vm + (size_t)s_codes[w4 + 2] * DG;
  const float* __restrict__ r3 = vm + (size_t)s_codes[w4 + 3] * DG;

  // out[((n*512 + m*64 + c) * 64 + h) * 64 + w]
  float* __restrict__ outbase =
      out + ((((size_t)n * 512 + (size_t)m * 64) * HW + h) * HW);

  #pragma unroll
  for (int p = 0; p < 4; ++p) {
    const int c = c0 + 16 * p;
    float4 val = make_float4(r0[c], r1[c], r2[c], r3[c]);
    *reinterpret_cast<float4*>(outbase + (size_t)c * (HW * HW) + w4) = val;
  }
}

extern "C" void kernel_launch(void* const* d_in, const int* in_sizes, int n_in,
                              void* d_out, int out_size, void* d_ws, size_t ws_size,
                              hipStream_t stream) {
  const int*   codes    = (const int*)d_in[0];
  const float* codebook = (const float*)d_in[1];
  const float* wv       = (const float*)d_in[2];
  float*       out      = (float*)d_out;
  float*       v        = (float*)d_ws;   // needs 8*4096*64*4 = 8 MB

  (void)in_sizes; (void)n_in; (void)out_size; (void)ws_size;

  qd_gemm_wmma<<<1024, 256, 0, stream>>>(codebook, wv, v);
  qd_gather<<<NB * MDIM * HW, 256, 0, stream>>>(codes, v, out);
}